// Linear_46866683134400
// MI455X (gfx1250) — compile-verified
//
#include <hip/hip_runtime.h>

#define B_DIM 16384
#define S_DIM 256
#define NCH   8
#define CH2   16
#define EPS   1e-5f

typedef float v2f __attribute__((ext_vector_type(2)));
typedef float v8f __attribute__((ext_vector_type(8)));

// workspace layout (floats)
#define WS_SUM   0
#define WS_SQ    16
#define WS_SCALE 32
#define WS_SHIFT 48
#define WS_Y     64              // 4 floats: s0,s1,q0,q1
#define WS_CR    256
#define WS_CI    (256 + B_DIM)
#define WS_TOTAL (256 + 2 * B_DIM)

#define BM  32                   // batch rows per block
#define LDA 260                  // padded LDS row stride (260 % 64 == 4 -> no DS bank conflicts)

__global__ void init_ws(float* __restrict__ ws) {
    int i = blockIdx.x * blockDim.x + threadIdx.x;
    if (i < WS_TOTAL) ws[i] = 0.f;
}

__global__ void __launch_bounds__(256) stats1(const float* __restrict__ x, float* __restrict__ ws) {
    __shared__ float sS[256];
    __shared__ float sQ[256];
    const int tid = threadIdx.x;
    float s = 0.f, q = 0.f;
    const int N = B_DIM * S_DIM * CH2;
    for (int i = blockIdx.x * 256 + tid; i < N; i += gridDim.x * 256) {
        float v = x[i];
        s += v;
        q += v * v;
    }
    sS[tid] = s; sQ[tid] = q;
    __syncthreads();
    if (tid < CH2) {
        float a = 0.f, b = 0.f;
#pragma unroll
        for (int j = 0; j < 16; ++j) { a += sS[tid + 16 * j]; b += sQ[tid + 16 * j]; }
        atomicAdd(ws + WS_SUM + tid, a);
        atomicAdd(ws + WS_SQ  + tid, b);
    }
}

__global__ void stats2(const float* __restrict__ g_in, const float* __restrict__ b_in,
                       float* __restrict__ ws) {
    int c = threadIdx.x;
    if (c < CH2) {
        const float invN = 1.f / ((float)B_DIM * (float)S_DIM);
        float mu  = ws[WS_SUM + c] * invN;
        float var = ws[WS_SQ  + c] * invN - mu * mu;
        float a   = g_in[c] * rsqrtf(var + EPS);
        ws[WS_SCALE + c] = a;
        ws[WS_SHIFT + c] = b_in[c] - mu * a;
    }
}

__device__ __forceinline__ v8f wmma4(v2f a, v2f b, v8f c) {
    // D = A(16x4 f32) x B(4x16 f32) + C, full fp32
    return __builtin_amdgcn_wmma_f32_16x16x4_f32(false, a, false, b, (short)0, c, false, false);
}

__global__ void __launch_bounds__(256)
gemm_fused(const float* __restrict__ x, const float* __restrict__ Wr,
           const float* __restrict__ Wi, const float* __restrict__ Wfr,
           const float* __restrict__ Wfi, float* __restrict__ ws) {
    extern __shared__ float smem[];
    float* sAi = smem;             // [BM][LDA] normalized in-phase plane
    float* sAq = smem + BM * LDA;  // [BM][LDA] normalized quadrature plane

    const int tid   = threadIdx.x;
    const int lane  = tid & 31;
    const int w     = tid >> 5;
    const int waveM = w >> 2;            // 0..1 : 16-row strip
    const int waveN = w & 3;             // 0..3 : 16-col strip
    const int m     = lane & 15;         // A row / B col / C col within tile
    const int kh    = lane >> 4;         // k-half select
    const int b0    = blockIdx.x * BM;

    const float2* xf2 = (const float2*)x;
    float* CrAcc = ws + WS_CR;
    float* CiAcc = ws + WS_CI;
    const int rowBase = waveM * 16;

    for (int c = 0; c < NCH; ++c) {
        const float scI = ws[WS_SCALE + 2 * c];
        const float shI = ws[WS_SHIFT + 2 * c];
        const float scQ = ws[WS_SCALE + 2 * c + 1];
        const float shQ = ws[WS_SHIFT + 2 * c + 1];

        __syncthreads();   // protect previous channel's LDS reads
        // Stage normalized A planes: rows b0..b0+BM-1, s == tid (full 0..255)
        for (int r = 0; r < BM; ++r) {
            int bs = (b0 + r) * S_DIM + tid;
            float2 v = xf2[bs * NCH + c];   // x[b, s, 2c] and x[b, s, 2c+1]
            sAi[r * LDA + tid] = v.x * scI + shI;
            sAq[r * LDA + tid] = v.y * scQ + shQ;
        }
        __syncthreads();

        const float* WrC = Wr + c * (S_DIM * S_DIM);
        const float* WiC = Wi + c * (S_DIM * S_DIM);

        for (int niter = 0; niter < 4; ++niter) {
            const int o = niter * 64 + waveN * 16 + m;   // output column for this lane
            const float* bRp = WrC + o * S_DIM + 2 * kh; // WrT[k, o] = Wr[c, o, k]
            const float* bIp = WiC + o * S_DIM + 2 * kh;
            const float* aIp = sAi + (rowBase + m) * LDA + 2 * kh;
            const float* aQp = sAq + (rowBase + m) * LDA + 2 * kh;

            v8f p1 = {0.f,0.f,0.f,0.f,0.f,0.f,0.f,0.f};  // xi . Wr
            v8f p2 = {0.f,0.f,0.f,0.f,0.f,0.f,0.f,0.f};  // xq . Wi
            v8f p3 = {0.f,0.f,0.f,0.f,0.f,0.f,0.f,0.f};  // xq . Wr
            v8f p4 = {0.f,0.f,0.f,0.f,0.f,0.f,0.f,0.f};  // xi . Wi

#pragma unroll 4
            for (int k0 = 0; k0 < S_DIM; k0 += 4) {
                v2f ai = *(const v2f*)(aIp + k0);
                v2f aq = *(const v2f*)(aQp + k0);
                v2f br = *(const v2f*)(bRp + k0);
                v2f bi = *(const v2f*)(bIp + k0);
                p1 = wmma4(ai, br, p1);
                p2 = wmma4(aq, bi, p2);
                p3 = wmma4(aq, br, p3);
                p4 = wmma4(ai, bi, p4);
            }

            // Fused epilogue: amp2 * (cr,ci) projected onto Wfr/Wfi, reduced over o
            const float wfr = Wfr[c * S_DIM + o];
            const float wfi = Wfi[c * S_DIM + o];
#pragma unroll
            for (int v = 0; v < 8; ++v) {
                const int rloc = rowBase + 8 * kh + v;   // C layout: lanes>=16 hold M+8
                float cr = p1[v] - p2[v];
                float ci = p3[v] + p4[v];
                float ai = sAi[rloc * LDA + o];
                float aq = sAq[rloc * LDA + o];
                float amp2 = ai * ai + aq * aq;
                float tCr = amp2 * (cr * wfr - ci * wfi);
                float tCi = amp2 * (ci * wfr + cr * wfi);
#pragma unroll
                for (int sft = 1; sft < 16; sft <<= 1) {
                    tCr += __shfl_xor(tCr, sft, 32);
                    tCi += __shfl_xor(tCi, sft, 32);
                }
                if (m == 0) {
                    atomicAdd(&CrAcc[b0 + rloc], tCr);
                    atomicAdd(&CiAcc[b0 + rloc], tCi);
                }
            }
        }
    }
}

__global__ void __launch_bounds__(256) ynorm1(float* __restrict__ ws) {
    __shared__ float r0[256], r1[256], r2[256], r3[256];
    const int tid = threadIdx.x;
    float s0 = 0.f, s1 = 0.f, q0 = 0.f, q1 = 0.f;
    for (int b = blockIdx.x * 256 + tid; b < B_DIM; b += gridDim.x * 256) {
        float v0 = ws[WS_CR + b], v1 = ws[WS_CI + b];
        s0 += v0; s1 += v1; q0 += v0 * v0; q1 += v1 * v1;
    }
    r0[tid] = s0; r1[tid] = s1; r2[tid] = q0; r3[tid] = q1;
    __syncthreads();
    for (int st = 128; st > 0; st >>= 1) {
        if (tid < st) {
            r0[tid] += r0[tid + st]; r1[tid] += r1[tid + st];
            r2[tid] += r2[tid + st]; r3[tid] += r3[tid + st];
        }
        __syncthreads();
    }
    if (tid == 0) {
        atomicAdd(ws + WS_Y + 0, r0[0]);
        atomicAdd(ws + WS_Y + 1, r1[0]);
        atomicAdd(ws + WS_Y + 2, r2[0]);
        atomicAdd(ws + WS_Y + 3, r3[0]);
    }
}

__global__ void ynorm2(const float* __restrict__ ws, const float* __restrict__ g_out,
                       const float* __restrict__ b_out, float* __restrict__ out) {
    int b = blockIdx.x * blockDim.x + threadIdx.x;
    if (b >= B_DIM) return;
    const float invB = 1.f / (float)B_DIM;
    float mu0 = ws[WS_Y + 0] * invB, mu1 = ws[WS_Y + 1] * invB;
    float v0  = ws[WS_Y + 2] * invB - mu0 * mu0;
    float v1  = ws[WS_Y + 3] * invB - mu1 * mu1;
    float rs0 = rsqrtf(v0 + EPS), rs1 = rsqrtf(v1 + EPS);
    float y0 = ws[WS_CR + b], y1 = ws[WS_CI + b];
    out[2 * b + 0] = (y0 - mu0) * rs0 * g_out[0] + b_out[0];
    out[2 * b + 1] = (y1 - mu1) * rs1 * g_out[1] + b_out[1];
}

extern "C" void kernel_launch(void* const* d_in, const int* in_sizes, int n_in,
                              void* d_out, int out_size, void* d_ws, size_t ws_size,
                              hipStream_t stream) {
    (void)in_sizes; (void)n_in; (void)out_size; (void)ws_size;
    const float* x     = (const float*)d_in[0];
    // d_in[1] = h_0 (unused by the reference)
    const float* Wr    = (const float*)d_in[2];
    const float* Wi    = (const float*)d_in[3];
    const float* Wfr   = (const float*)d_in[4];
    const float* Wfi   = (const float*)d_in[5];
    const float* g_in  = (const float*)d_in[6];
    const float* b_in  = (const float*)d_in[7];
    const float* g_out = (const float*)d_in[8];
    const float* b_out = (const float*)d_in[9];
    float* ws  = (float*)d_ws;
    float* out = (float*)d_out;

    init_ws<<<(WS_TOTAL + 255) / 256, 256, 0, stream>>>(ws);
    stats1<<<2048, 256, 0, stream>>>(x, ws);
    stats2<<<1, 32, 0, stream>>>(g_in, b_in, ws);

    const size_t smem = (size_t)2 * BM * LDA * sizeof(float);  // 66,560 B
    gemm_fused<<<B_DIM / BM, 256, smem, stream>>>(x, Wr, Wi, Wfr, Wfi, ws);

    ynorm1<<<64, 256, 0, stream>>>(ws);
    ynorm2<<<(B_DIM + 255) / 256, 256, 0, stream>>>(ws, g_out, b_out, out);
}